// APPNP_net_lr_84954453115010
// MI455X (gfx1250) — compile-verified
//
#include <hip/hip_runtime.h>
#include <math.h>

// ---------------------------------------------------------------------------
// Problem constants (from the reference)
// ---------------------------------------------------------------------------
#define NN     100000
#define EE     1600000
#define IN_C   256
#define HID    64
#define OUT_C  64
#define KPROP  10
#define ALPHA  0.1f
#define LN_EPS 1e-5f

typedef float v2f  __attribute__((ext_vector_type(2)));
typedef float v8f  __attribute__((ext_vector_type(8)));
typedef unsigned int u32x4 __attribute__((ext_vector_type(4)));
typedef int   i32x8 __attribute__((ext_vector_type(8)));
typedef int   i32x4 __attribute__((ext_vector_type(4)));

__device__ __forceinline__ float gelu_exact(float x) {
    // torch-default exact GELU: 0.5*x*(1+erf(x/sqrt(2)))
    return 0.5f * x * (1.0f + erff(x * 0.70710678118654752440f));
}

// LDS row stride for the x tile: 256 data DWORDs + 4 pad DWORDs (TDM pad),
// 260 % 64 == 4 -> the 16 A-rows read by lanes 0..15 hit 16 distinct banks.
#define XT_STRIDE 260

// ---------------------------------------------------------------------------
// Kernel 1: h0 = LayerNorm(gelu(x @ W1 + b1))
// Block = 128 threads = 4 wave32; block computes 16 rows x 64 cols.
// A tile (16x256 f32, 16 KB) is DMA'd into LDS by the Tensor Data Mover in a
// single descriptor (pad_enable inserts 4 DWORDs per 256 for bank spread);
// the WMMA K-loop then reads A via ds_load and B (W1, L2-hot) from global.
// ---------------------------------------------------------------------------
__global__ void __launch_bounds__(128)
gemm1_bias_gelu_ln(const float* __restrict__ x,  const float* __restrict__ W1,
                   const float* __restrict__ b1, const float* __restrict__ g1,
                   const float* __restrict__ be1, float* __restrict__ h0) {
    __shared__ float lds_x[16 * XT_STRIDE];           // 16.6 KB
    __shared__ float tile[16][HID + 4];
    __shared__ float redS[16][8], redQ[16][8], mu[16], rstd[16];

    const int tid  = threadIdx.x;
    const int row0 = blockIdx.x * 16;

#if __has_builtin(__builtin_amdgcn_tensor_load_to_lds) && __has_builtin(__builtin_amdgcn_s_wait_tensorcnt)
    if (tid < 32) {  // one wave issues the TDM DMA (EXEC is ignored by TDM)
        const unsigned long long ga =
            (unsigned long long)(const void*)(x + (size_t)row0 * IN_C);
        const unsigned lds_off =
            (unsigned)(unsigned long long)(uintptr_t)(void*)&lds_x[0]; // addr[31:0] = LDS offset

        u32x4 g0;
        g0[0] = 1u;                                          // count=1, no gather
        g0[1] = lds_off;                                     // lds_addr
        g0[2] = (unsigned)(ga & 0xffffffffu);                // global_addr[31:0]
        g0[3] = (unsigned)((ga >> 32) & 0x01ffffffu)         // global_addr[56:32]
              | (2u << 30);                                  // type=2 (image)

        i32x8 g1v;
        g1v[0] = (int)((2u << 16)      // data_size = 4 B
                     | (1u << 20)      // pad_enable
                     | (7u << 22)      // pad_interval: 256 DWORDs
                     | (3u << 25));    // pad_amount:   4 DWORDs
        g1v[1] = (int)(IN_C << 16);    // tensor_dim0[15:0]=256 in bits [31:16]
        g1v[2] = (int)(16u  << 16);    // tensor_dim0 hi=0 | tensor_dim1 lo=16
        g1v[3] = (int)(IN_C << 16);    // tensor_dim1 hi=0 | tile_dim0=256
        g1v[4] = 16;                   // tile_dim1=16, tile_dim2=0
        g1v[5] = IN_C;                 // tensor_dim0_stride[31:0] = 256
        g1v[6] = 0;                    // stride0 hi | tensor_dim1_stride lo
        g1v[7] = 0;
        const i32x4 z4 = {0, 0, 0, 0};               // groups 2/3 unused (2-D)
        const i32x8 z8 = {0, 0, 0, 0, 0, 0, 0, 0};   // trailing group unused
        __builtin_amdgcn_tensor_load_to_lds(g0, g1v, z4, z4, z8, 0);
        __builtin_amdgcn_s_wait_tensorcnt(0);
    }
    __syncthreads();
#else
    // Fallback: cooperative staging with the same padded layout.
    for (int i = tid; i < 16 * IN_C; i += 128) {
        const int r = i >> 8;
        const int k = i & 255;
        lds_x[r * XT_STRIDE + k] = x[(size_t)(row0 + r) * IN_C + k];
    }
    __syncthreads();
#endif

    const int wave = tid >> 5;        // 0..3 -> N tile
    const int lane = tid & 31;
    const int mr   = lane & 15;       // A row within tile == B col within tile
    const int half = lane >> 4;       // 0/1 -> K sub-slice
    const float* arow = &lds_x[mr * XT_STRIDE];
    const float* bcol = W1 + (16 * wave + mr);

    v8f acc = {};
    for (int k0 = 0; k0 < IN_C; k0 += 4) {
        v2f a, b;
        a.x = arow[k0 + 2 * half];                     // A[mr][k0+2*half]
        a.y = arow[k0 + 2 * half + 1];
        b.x = bcol[(size_t)(k0 + 2 * half)     * HID]; // B[k0+2*half][col]
        b.y = bcol[(size_t)(k0 + 2 * half + 1) * HID];
        acc = __builtin_amdgcn_wmma_f32_16x16x4_f32(false, a, false, b,
                                                    (short)0, acc, false, false);
    }

    // Epilogue: bias + gelu into LDS tile, then per-row LayerNorm over HID=64.
    const int ncol = 16 * wave + mr;
    #pragma unroll
    for (int v = 0; v < 8; v++) {
        const int r = v + 8 * half;                   // D row within tile
        tile[r][ncol] = gelu_exact(acc[v] + b1[ncol]);
    }
    __syncthreads();

    const int r   = tid >> 3;   // 0..15: row
    const int sub = tid & 7;    // 8 threads per row, 8 cols each
    float s = 0.f, q = 0.f;
    #pragma unroll
    for (int i = 0; i < 8; i++) {
        const float t = tile[r][sub * 8 + i];
        s += t; q += t * t;
    }
    redS[r][sub] = s; redQ[r][sub] = q;
    __syncthreads();
    if (sub == 0) {
        float S = 0.f, Q = 0.f;
        #pragma unroll
        for (int i = 0; i < 8; i++) { S += redS[r][i]; Q += redQ[r][i]; }
        const float mean = S * (1.0f / HID);
        const float var  = Q * (1.0f / HID) - mean * mean;
        mu[r]   = mean;
        rstd[r] = rsqrtf(var + LN_EPS);
    }
    __syncthreads();
    float* orow = h0 + (size_t)(row0 + r) * HID;
    #pragma unroll
    for (int i = 0; i < 8; i++) {
        const int c = sub * 8 + i;
        orow[c] = (tile[r][c] - mu[r]) * rstd[r] * g1[c] + be1[c];
    }
}

// ---------------------------------------------------------------------------
// CSR build for gcn_norm(edge_index, add_self_loops): deg[c] = indeg(c)+1,
// dinv = rsqrt(deg), edges grouped by destination, packed {src, w} records.
// ---------------------------------------------------------------------------
struct __align__(8) EdgeRec { int src; float w; };

__global__ void count_deg(const int* __restrict__ col, int* __restrict__ cnt) {
    const int e = blockIdx.x * blockDim.x + threadIdx.x;
    if (e < EE) atomicAdd(&cnt[col[e]], 1);
}

__global__ void make_dinv(const int* __restrict__ cnt, float* __restrict__ dinv) {
    const int i = blockIdx.x * blockDim.x + threadIdx.x;
    if (i < NN) dinv[i] = rsqrtf((float)(cnt[i] + 1));  // +1 = self loop; always > 0
}

// Single-block exclusive scan over NN counts -> rowptr[NN+1].
__global__ void __launch_bounds__(1024)
scan_rowptr(const int* __restrict__ cnt, int* __restrict__ rowptr) {
    __shared__ int sums[1024];
    const int tid   = threadIdx.x;
    const int chunk = (NN + 1023) / 1024;
    const int start = tid * chunk;
    const int end   = (start + chunk < NN) ? (start + chunk) : NN;
    int s = 0;
    for (int i = start; i < end; i++) s += cnt[i];
    sums[tid] = s;
    __syncthreads();
    if (tid == 0) {
        int acc = 0;
        for (int i = 0; i < 1024; i++) { const int t = sums[i]; sums[i] = acc; acc += t; }
        rowptr[NN] = acc;   // == EE
    }
    __syncthreads();
    int acc = sums[tid];
    for (int i = start; i < end; i++) { rowptr[i] = acc; acc += cnt[i]; }
}

__global__ void scatter_csr(const int* __restrict__ rowv, const int* __restrict__ colv,
                            const float* __restrict__ dinv, const int* __restrict__ rowptr,
                            int* __restrict__ cursor, EdgeRec* __restrict__ csr) {
    const int e = blockIdx.x * blockDim.x + threadIdx.x;
    if (e < EE) {
        const int r = rowv[e];
        const int c = colv[e];
        const int slot = rowptr[c] + atomicAdd(&cursor[c], 1);
        EdgeRec rec; rec.src = r; rec.w = dinv[r] * dinv[c];
        csr[slot] = rec;
    }
}

// ---------------------------------------------------------------------------
// One APPNP round: hout = (1-a) * (A_hat @ hin) + a * h0.
// One wave32 per destination node, float2 per lane (64 floats = 32 x 8 B),
// 4 nodes per 128-thread block. Per edge: one broadcast b64 (packed record)
// + one b64 row-gather per lane, unrolled x2 for memory-level parallelism.
// h (25.6 MB) + CSR (12.8 MB) stay resident in MI455X's 192 MB L2.
// ---------------------------------------------------------------------------
__global__ void __launch_bounds__(128)
appnp_step(const float* __restrict__ hin, const float* __restrict__ h0,
           const float* __restrict__ dinv, const int* __restrict__ rowptr,
           const EdgeRec* __restrict__ csr, float* __restrict__ hout) {
    const int wave = threadIdx.x >> 5;
    const int lane = threadIdx.x & 31;
    const int c    = blockIdx.x * 4 + wave;           // destination node
    const int beg  = rowptr[c];
    const int end  = rowptr[c + 1];
    const float di = dinv[c];

    const size_t foff = 2 * lane;                     // this lane's feature pair
    float2 acc = *(const float2*)(hin + (size_t)c * HID + foff);
    acc.x *= di * di;                                 // self-loop term
    acc.y *= di * di;

    int e = beg;
    for (; e + 2 <= end; e += 2) {
        const EdgeRec r0 = csr[e];
        const EdgeRec r1 = csr[e + 1];
        const float2 a0 = *(const float2*)(hin + (size_t)r0.src * HID + foff);
        const float2 a1 = *(const float2*)(hin + (size_t)r1.src * HID + foff);
        acc.x = fmaf(r0.w, a0.x, acc.x);
        acc.y = fmaf(r0.w, a0.y, acc.y);
        acc.x = fmaf(r1.w, a1.x, acc.x);
        acc.y = fmaf(r1.w, a1.y, acc.y);
    }
    if (e < end) {
        const EdgeRec r0 = csr[e];
        const float2 a0 = *(const float2*)(hin + (size_t)r0.src * HID + foff);
        acc.x = fmaf(r0.w, a0.x, acc.x);
        acc.y = fmaf(r0.w, a0.y, acc.y);
    }

    const float2 h0v = *(const float2*)(h0 + (size_t)c * HID + foff);
    float2 outv;
    outv.x = (1.0f - ALPHA) * acc.x + ALPHA * h0v.x;
    outv.y = (1.0f - ALPHA) * acc.y + ALPHA * h0v.y;
    *(float2*)(hout + (size_t)c * HID + foff) = outv;
}

// ---------------------------------------------------------------------------
// Kernel 3: out = LayerNorm(gelu(hK)) @ W2 + b2
// GELU+LN done in LDS on the 16x64 A tile (stride 68 -> conflict-free),
// then WMMA f32 with K=64.
// ---------------------------------------------------------------------------
__global__ void __launch_bounds__(128)
gemm2_out(const float* __restrict__ hK, const float* __restrict__ g2,
          const float* __restrict__ be2, const float* __restrict__ W2,
          const float* __restrict__ b2, float* __restrict__ out) {
    __shared__ float tile[16][HID + 4];
    __shared__ float redS[16][8], redQ[16][8], mu[16], rstd[16];

    const int tid  = threadIdx.x;
    const int row0 = blockIdx.x * 16;

    // Stage A = LN(gelu(hK tile)) in LDS. 8 threads per row, 8 cols each.
    {
        const int r   = tid >> 3;
        const int sub = tid & 7;
        const float* hrow = hK + (size_t)(row0 + r) * HID;
        float vloc[8];
        float s = 0.f, q = 0.f;
        #pragma unroll
        for (int i = 0; i < 8; i++) {
            const float t = gelu_exact(hrow[sub * 8 + i]);
            vloc[i] = t; s += t; q += t * t;
        }
        redS[r][sub] = s; redQ[r][sub] = q;
        __syncthreads();
        if (sub == 0) {
            float S = 0.f, Q = 0.f;
            #pragma unroll
            for (int i = 0; i < 8; i++) { S += redS[r][i]; Q += redQ[r][i]; }
            const float mean = S * (1.0f / HID);
            const float var  = Q * (1.0f / HID) - mean * mean;
            mu[r]   = mean;
            rstd[r] = rsqrtf(var + LN_EPS);
        }
        __syncthreads();
        #pragma unroll
        for (int i = 0; i < 8; i++) {
            const int c = sub * 8 + i;
            tile[r][c] = (vloc[i] - mu[r]) * rstd[r] * g2[c] + be2[c];
        }
        __syncthreads();
    }

    // WMMA: D = tile(16x64) @ W2(64x64) + b2, wave w owns cols [16w,16w+16)
    const int wave = tid >> 5;
    const int lane = tid & 31;
    const int mr   = lane & 15;
    const int half = lane >> 4;
    const float* bcol = W2 + (16 * wave + mr);

    v8f acc = {};
    #pragma unroll
    for (int k0 = 0; k0 < HID; k0 += 4) {
        v2f a, b;
        a.x = tile[mr][k0 + 2 * half];
        a.y = tile[mr][k0 + 2 * half + 1];
        b.x = bcol[(size_t)(k0 + 2 * half)     * OUT_C];
        b.y = bcol[(size_t)(k0 + 2 * half + 1) * OUT_C];
        acc = __builtin_amdgcn_wmma_f32_16x16x4_f32(false, a, false, b,
                                                    (short)0, acc, false, false);
    }

    const int ncol = 16 * wave + mr;
    #pragma unroll
    for (int v = 0; v < 8; v++) {
        const int r = v + 8 * half;
        out[(size_t)(row0 + r) * OUT_C + ncol] = acc[v] + b2[ncol];
    }
}

// ---------------------------------------------------------------------------
// Host-side launcher
// ---------------------------------------------------------------------------
static inline char* ws_bump(char*& p, size_t bytes) {
    char* r = p;
    p += (bytes + 255) & ~(size_t)255;
    return r;
}

extern "C" void kernel_launch(void* const* d_in, const int* in_sizes, int n_in,
                              void* d_out, int out_size, void* d_ws, size_t ws_size,
                              hipStream_t stream) {
    (void)in_sizes; (void)n_in; (void)out_size; (void)ws_size;

    const float* x   = (const float*)d_in[0];
    const int*   ei  = (const int*)  d_in[1];   // [2, E] flat: rows then cols
    const float* W1  = (const float*)d_in[2];
    const float* b1  = (const float*)d_in[3];
    const float* g1  = (const float*)d_in[4];
    const float* be1 = (const float*)d_in[5];
    const float* g2  = (const float*)d_in[6];
    const float* be2 = (const float*)d_in[7];
    const float* W2  = (const float*)d_in[8];
    const float* b2  = (const float*)d_in[9];
    const int* erow = ei;
    const int* ecol = ei + EE;

    // Workspace carve-out (~91 MB total)
    char* wp = (char*)d_ws;
    float*   h0     = (float*)  ws_bump(wp, (size_t)NN * HID * sizeof(float));
    float*   hA     = (float*)  ws_bump(wp, (size_t)NN * HID * sizeof(float));
    float*   hB     = (float*)  ws_bump(wp, (size_t)NN * HID * sizeof(float));
    int*     cnt    = (int*)    ws_bump(wp, (size_t)NN * sizeof(int));
    float*   dinv   = (float*)  ws_bump(wp, (size_t)NN * sizeof(float));
    int*     rowptr = (int*)    ws_bump(wp, (size_t)(NN + 1) * sizeof(int));
    int*     cursor = (int*)    ws_bump(wp, (size_t)NN * sizeof(int));
    EdgeRec* csr    = (EdgeRec*)ws_bump(wp, (size_t)EE * sizeof(EdgeRec));

    (void)hipMemsetAsync(cnt,    0, (size_t)NN * sizeof(int), stream);
    (void)hipMemsetAsync(cursor, 0, (size_t)NN * sizeof(int), stream);

    // 1) h0 = LN(gelu(x @ W1 + b1))  — TDM tile load + WMMA fp32
    gemm1_bias_gelu_ln<<<NN / 16, 128, 0, stream>>>(x, W1, b1, g1, be1, h0);

    // 2) Build destination-grouped CSR + symmetric normalization
    count_deg  <<<(EE + 255) / 256, 256, 0, stream>>>(ecol, cnt);
    make_dinv  <<<(NN + 255) / 256, 256, 0, stream>>>(cnt, dinv);
    scan_rowptr<<<1, 1024, 0, stream>>>(cnt, rowptr);
    scatter_csr<<<(EE + 255) / 256, 256, 0, stream>>>(erow, ecol, dinv, rowptr,
                                                      cursor, csr);

    // 3) K=10 APPNP rounds, atomic-free gather, ping-pong buffers
    const float* hin = h0;
    float* hout = hA;
    for (int it = 0; it < KPROP; it++) {
        appnp_step<<<NN / 4, 128, 0, stream>>>(hin, h0, dinv, rowptr, csr, hout);
        hin  = hout;
        hout = (hout == hA) ? hB : hA;
    }

    // 4) out = LN(gelu(hK)) @ W2 + b2 — WMMA fp32, LN fused via LDS tile
    gemm2_out<<<NN / 16, 128, 0, stream>>>(hin, g2, be2, W2, b2, (float*)d_out);
}